// GCN_54443005444259
// MI455X (gfx1250) — compile-verified
//
#include <hip/hip_runtime.h>

typedef __attribute__((ext_vector_type(2))) float v2f;
typedef __attribute__((ext_vector_type(8))) float v8f;

#define HID 64
#define HID2 32
#define NLAYER 3
#define BN_EPS 1e-5f

// ---------------------------------------------------------------------------
// Native CDNA5 f32 atomic add (no-return form, device scope -> RMW at L2).
// Avoids any chance of a CAS-loop expansion from HIP's atomicAdd(float*).
// Tracked by STOREcnt; S_ENDPGM performs an implicit wait-idle.
// ---------------------------------------------------------------------------
__device__ __forceinline__ void atomic_add_f32(float* p, float v) {
  asm volatile("global_atomic_add_f32 %0, %1, off scope:SCOPE_DEV"
               :
               : "v"(p), "v"(v)
               : "memory");
}

// ---------------------------------------------------------------------------
// Degree / normalization
// ---------------------------------------------------------------------------
__global__ void k_deg_init(float* __restrict__ deg, int N) {
  int i = blockIdx.x * blockDim.x + threadIdx.x;
  if (i < N) deg[i] = 1.0f;                       // self-loop contributes 1
}

__global__ void k_deg_count(const int* __restrict__ dst, float* __restrict__ deg, int E) {
  int e = blockIdx.x * blockDim.x + threadIdx.x;
  if (e < E) atomic_add_f32(&deg[dst[e]], 1.0f);
}

__global__ void k_dinv(const float* __restrict__ deg, float* __restrict__ dinv, int N) {
  int i = blockIdx.x * blockDim.x + threadIdx.x;
  if (i < N) dinv[i] = rsqrtf(fmaxf(deg[i], 1.0f));
}

// ---------------------------------------------------------------------------
// WMMA fp32 GEMM:  C[M,64] = A[M,K] @ B[K,64] (+ optional bias)
// block = 128 threads (4 waves); wave w computes cols [16w,16w+16) of a
// 16-row tile.  Uses V_WMMA_F32_16X16X4_F32, K stepped by 4.
// ---------------------------------------------------------------------------
__global__ void k_gemm_wmma(const float* __restrict__ A, const float* __restrict__ B,
                            const float* __restrict__ bias, float* __restrict__ C,
                            int M, int K) {
  const int lane = threadIdx.x & 31;
  const int colBase = (threadIdx.x >> 5) * 16;
  const int row0 = blockIdx.x * 16;
  const int m = lane & 15;
  const int kh = (lane >> 4) << 1;               // 0 or 2
  int rowA = row0 + m;
  if (rowA >= M) rowA = M - 1;                   // clamp (stores are guarded)
  const float* __restrict__ Arow = A + (size_t)rowA * K;
  const int col = colBase + m;

  v8f c = {};
  for (int k0 = 0; k0 < K; k0 += 4) {
    v2f a, b;
    a.x = Arow[k0 + kh];
    a.y = Arow[k0 + kh + 1];
    b.x = B[(size_t)(k0 + kh) * HID + col];
    b.y = B[(size_t)(k0 + kh + 1) * HID + col];
    c = __builtin_amdgcn_wmma_f32_16x16x4_f32(false, a, false, b, (short)0, c,
                                              false, false);
  }

  const float bb = bias ? bias[col] : 0.0f;
  const int rbase = row0 + ((lane >> 4) << 3);
#pragma unroll
  for (int v = 0; v < 8; ++v) {
    int r = rbase + v;
    if (r < M) C[(size_t)r * HID + col] = c[v] + bb;
  }
}

// ---------------------------------------------------------------------------
// agg[i][c] = conv_b[c]   (bias pre-seeded so scatter just accumulates)
// ---------------------------------------------------------------------------
__global__ void k_agg_init(const float* __restrict__ bias, float* __restrict__ agg,
                           long long total) {
  long long t = (long long)blockIdx.x * blockDim.x + threadIdx.x;
  if (t < total) agg[t] = bias[t & (HID - 1)];
}

// ---------------------------------------------------------------------------
// Edge scatter: agg[dst] += hw[src] * dinv[src]*dinv[dst]
// 16 threads per (edge or self-loop), each handles 4 contiguous channels.
// float4 gather + 4 native f32 atomics.
// ---------------------------------------------------------------------------
__global__ void k_scatter(const int* __restrict__ src, const int* __restrict__ dst,
                          const float* __restrict__ dinv, const float* __restrict__ hw,
                          float* __restrict__ agg, int E, int N) {
  long long t = (long long)blockIdx.x * blockDim.x + threadIdx.x;
  long long e = t >> 4;
  if (e >= (long long)E + N) return;
  int c4 = ((int)t & 15) * 4;
  int s, d;
  if (e < E) { s = src[e]; d = dst[e]; }
  else       { s = d = (int)(e - E); }            // self loop
  float w = dinv[s] * dinv[d];
  const float4 v = *reinterpret_cast<const float4*>(hw + (size_t)s * HID + c4);
  float* ap = agg + (size_t)d * HID + c4;
  atomic_add_f32(ap + 0, v.x * w);
  atomic_add_f32(ap + 1, v.y * w);
  atomic_add_f32(ap + 2, v.z * w);
  atomic_add_f32(ap + 3, v.w * w);
}

// ---------------------------------------------------------------------------
// BatchNorm (running stats) + ReLU, writes next-layer h
// ---------------------------------------------------------------------------
__global__ void k_bn_relu(const float* __restrict__ agg, const float* __restrict__ gamma,
                          const float* __restrict__ beta, const float* __restrict__ mean,
                          const float* __restrict__ var, float* __restrict__ h,
                          long long total) {
  long long t = (long long)blockIdx.x * blockDim.x + threadIdx.x;
  if (t >= total) return;
  int c = (int)t & (HID - 1);
  float v = (agg[t] - mean[c]) * rsqrtf(var[c] + BN_EPS) * gamma[c] + beta[c];
  h[t] = v > 0.0f ? v : 0.0f;
}

// ---------------------------------------------------------------------------
// Global mean pool
// ---------------------------------------------------------------------------
__global__ void k_pool_zero(float* __restrict__ psum, float* __restrict__ pcnt, int G) {
  int t = blockIdx.x * blockDim.x + threadIdx.x;
  if (t < G * HID) psum[t] = 0.0f;
  if (t < G) pcnt[t] = 0.0f;
}

__global__ void k_pool_cnt(const int* __restrict__ batch, float* __restrict__ pcnt, int N) {
  int i = blockIdx.x * blockDim.x + threadIdx.x;
  if (i < N) atomic_add_f32(&pcnt[batch[i]], 1.0f);
}

__global__ void k_pool_acc(const int* __restrict__ batch, const float* __restrict__ h,
                           float* __restrict__ psum, int N) {
  long long t = (long long)blockIdx.x * blockDim.x + threadIdx.x;
  long long i = t >> 4;
  if (i >= N) return;
  int c4 = ((int)t & 15) * 4;
  int b = batch[i];
  const float4 v = *reinterpret_cast<const float4*>(h + (size_t)i * HID + c4);
  float* ap = psum + (size_t)b * HID + c4;
  atomic_add_f32(ap + 0, v.x);
  atomic_add_f32(ap + 1, v.y);
  atomic_add_f32(ap + 2, v.z);
  atomic_add_f32(ap + 3, v.w);
}

__global__ void k_pool_div(float* __restrict__ psum, const float* __restrict__ pcnt, int G) {
  int t = blockIdx.x * blockDim.x + threadIdx.x;
  if (t < G * HID) psum[t] /= fmaxf(pcnt[t >> 6], 1.0f);
}

// ---------------------------------------------------------------------------
// Tiny readout MLP
// ---------------------------------------------------------------------------
__global__ void k_mlp1(const float* __restrict__ g, const float* __restrict__ W1,
                       const float* __restrict__ b1, float* __restrict__ g1, int G) {
  int t = blockIdx.x * blockDim.x + threadIdx.x;
  if (t >= G * HID) return;
  int gi = t >> 6, j = t & (HID - 1);
  float acc = b1[j];
  for (int k = 0; k < HID; ++k) acc += g[gi * HID + k] * W1[k * HID + j];
  g1[t] = acc > 0.0f ? acc : 0.0f;
}

__global__ void k_mlp2(const float* __restrict__ g1, const float* __restrict__ W2,
                       const float* __restrict__ b2, float* __restrict__ g2, int G) {
  int t = blockIdx.x * blockDim.x + threadIdx.x;
  if (t >= G * HID2) return;
  int gi = t >> 5, j = t & (HID2 - 1);
  float acc = b2[j];
  for (int k = 0; k < HID; ++k) acc += g1[gi * HID + k] * W2[k * HID2 + j];
  g2[t] = acc > 0.0f ? acc : 0.0f;
}

__global__ void k_mlp3(const float* __restrict__ g2, const float* __restrict__ W3,
                       const float* __restrict__ b3, float* __restrict__ out, int G) {
  int t = blockIdx.x * blockDim.x + threadIdx.x;
  if (t >= G) return;
  float acc = b3[0];
  for (int k = 0; k < HID2; ++k) acc += g2[t * HID2 + k] * W3[k];
  out[t] = acc;
}

// ---------------------------------------------------------------------------
// Launch
// ---------------------------------------------------------------------------
extern "C" void kernel_launch(void* const* d_in, const int* in_sizes, int n_in,
                              void* d_out, int out_size, void* d_ws, size_t ws_size,
                              hipStream_t stream) {
  const float* x       = (const float*)d_in[0];
  const int*   edge    = (const int*)d_in[1];
  const int*   batch   = (const int*)d_in[2];
  const float* W_in    = (const float*)d_in[3];
  const float* b_in    = (const float*)d_in[4];
  const float* conv_W  = (const float*)d_in[5];
  const float* conv_b  = (const float*)d_in[6];
  const float* bn_g    = (const float*)d_in[7];
  const float* bn_b    = (const float*)d_in[8];
  const float* bn_m    = (const float*)d_in[9];
  const float* bn_v    = (const float*)d_in[10];
  const float* W1      = (const float*)d_in[11];
  const float* b1      = (const float*)d_in[12];
  const float* W2      = (const float*)d_in[13];
  const float* b2      = (const float*)d_in[14];
  const float* W3      = (const float*)d_in[15];
  const float* b3      = (const float*)d_in[16];
  float* out = (float*)d_out;

  const int N = in_sizes[2];
  const int E = in_sizes[1] / 2;
  const int K_IN = in_sizes[0] / N;               // 128
  const int G = out_size;                         // 256

  const int* src = edge;
  const int* dst = edge + E;

  // workspace partition (floats)
  float* ws   = (float*)d_ws;
  float* deg  = ws;
  float* dinv = deg + N;
  float* h    = dinv + N;
  float* hw   = h + (size_t)N * HID;
  float* agg  = hw + (size_t)N * HID;
  float* psum = agg + (size_t)N * HID;
  float* pcnt = psum + (size_t)G * HID;
  float* g1   = pcnt + G;
  float* g2   = g1 + (size_t)G * HID;

  const int TB = 256;
  const long long NH = (long long)N * HID;
  const int gridNH = (int)((NH + TB - 1) / TB);
  const int gridN  = (N + TB - 1) / TB;
  const int gridE  = (E + TB - 1) / TB;
  const int gridTiles = (N + 15) / 16;
  const long long scatterT = ((long long)E + N) * 16;
  const int gridScatter = (int)((scatterT + TB - 1) / TB);
  const int gridPoolAcc = (int)(((long long)N * 16 + TB - 1) / TB);

  // degree + symmetric normalization
  k_deg_init<<<gridN, TB, 0, stream>>>(deg, N);
  k_deg_count<<<gridE, TB, 0, stream>>>(dst, deg, E);
  k_dinv<<<gridN, TB, 0, stream>>>(deg, dinv, N);

  // input projection: h = x @ W_in + b_in
  k_gemm_wmma<<<gridTiles, 128, 0, stream>>>(x, W_in, b_in, h, N, K_IN);

  // GCN layers
  for (int l = 0; l < NLAYER; ++l) {
    k_agg_init<<<gridNH, TB, 0, stream>>>(conv_b + l * HID, agg, NH);
    k_gemm_wmma<<<gridTiles, 128, 0, stream>>>(h, conv_W + (size_t)l * HID * HID,
                                               nullptr, hw, N, HID);
    k_scatter<<<gridScatter, TB, 0, stream>>>(src, dst, dinv, hw, agg, E, N);
    k_bn_relu<<<gridNH, TB, 0, stream>>>(agg, bn_g + l * HID, bn_b + l * HID,
                                         bn_m + l * HID, bn_v + l * HID, h, NH);
  }

  // global mean pool
  k_pool_zero<<<(G * HID + TB - 1) / TB, TB, 0, stream>>>(psum, pcnt, G);
  k_pool_cnt<<<gridN, TB, 0, stream>>>(batch, pcnt, N);
  k_pool_acc<<<gridPoolAcc, TB, 0, stream>>>(batch, h, psum, N);
  k_pool_div<<<(G * HID + TB - 1) / TB, TB, 0, stream>>>(psum, pcnt, G);

  // readout MLP
  k_mlp1<<<(G * HID + TB - 1) / TB, TB, 0, stream>>>(psum, W1, b1, g1, G);
  k_mlp2<<<(G * HID2 + TB - 1) / TB, TB, 0, stream>>>(g1, W2, b2, g2, G);
  k_mlp3<<<(G + TB - 1) / TB, TB, 0, stream>>>(g2, W3, b3, out, G);
}